// MeshLoss_30442728194175
// MI455X (gfx1250) — compile-verified
//
#include <hip/hip_runtime.h>
#include <hip/hip_bf16.h>

// MI455X mesh loss (point->mesh min sq-dist), fp32 WMMA formulation.
//
// The six Ericson dot products d1..d6 and |ap|^2,|bp|^2,|cp|^2 all derive
// from three K=3 GEMMs Ga=a.p, Gb=b.p, Gc=c.p plus per-triangle constants:
//   d1 = (Gb-Ga) - ab.a   d3 = (Gb-Ga) - ab.b   d5 = (Gb-Ga) - ab.c
//   d2 = (Gc-Ga) - ac.a   d4 = (Gc-Ga) - ac.b   d6 = (Gc-Ga) - ac.c
//   |ap|^2 = |p|^2 - 2Ga + |a|^2   (etc.)
// Each wave: 16 points x 16 triangles per tile via V_WMMA_F32_16X16X4_F32
// (K=3 padded to 4), then a branch-free VALU epilogue per pair.

typedef __attribute__((ext_vector_type(2))) float v2f;
typedef __attribute__((ext_vector_type(8))) float v8f;

#define MESH_EPS 1e-12f

__device__ __forceinline__ float sdiv_guard(float n, float d) {
    float dd = (fabsf(d) < MESH_EPS) ? 1.0f : d;
    return n / dd;
}

// ---------------------------------------------------------------------------
// Kernel 1: pos[b][i][d] = sigmoid(vertices[b][d][i]); zero batch accumulators
// ---------------------------------------------------------------------------
__global__ void mesh_prep(const float* __restrict__ verts,
                          float* __restrict__ pos,
                          float* __restrict__ acc) {
    int i = blockIdx.x * blockDim.x + threadIdx.x;   // 0 .. 16383 = b*4096 + s
    if (i < 4) acc[i] = 0.0f;
    if (i >= 4 * 4096) return;
    int b = i >> 12;
    int s = i & 4095;
    #pragma unroll
    for (int d = 0; d < 3; ++d) {
        float x = verts[(b * 3 + d) * 4096 + s];
        pos[i * 3 + d] = 1.0f / (1.0f + expf(-x));
    }
}

// ---------------------------------------------------------------------------
// Kernel 2: main compute. 1024 waves = 4 batches x 256 point-tiles (16 pts).
// Each wave loops over 512 triangle tiles (16 tris), 3x WMMA f32 16x16x4.
// ---------------------------------------------------------------------------
__global__ void __launch_bounds__(256) mesh_main(const float* __restrict__ pos,
                                                 const float* __restrict__ pc,
                                                 const int* __restrict__ faces,
                                                 float* __restrict__ acc) {
    const int lane  = threadIdx.x & 31;
    const int wave  = threadIdx.x >> 5;
    const int gwave = blockIdx.x * 8 + wave;      // 0..1023
    const int b     = gwave >> 8;                 // batch 0..3
    const int pbase = (gwave & 255) * 16;         // point tile base

    const int  m  = lane & 15;
    const bool lo = (lane < 16);

    // This lane's point (SCALE == 1.0 so pts == pc)
    const float* P = pc + (((b << 12) + pbase + m) * 3);
    const float px = P[0], py = P[1], pz = P[2];

    // A operand (16x4, K=3 padded): VGPR0 = K0 / K2, VGPR1 = K1 / K3(=0)
    v2f A;
    A.x = lo ? px : pz;
    A.y = lo ? py : 0.0f;

    // |p|^2 for the 8 M-rows this lane owns in the C/D layout:
    // lanes<16 own M=r, lanes>=16 own M=r+8.
    const float q = px * px + py * py + pz * pz;
    float pp[8];
    #pragma unroll
    for (int r = 0; r < 8; ++r)
        pp[r] = __shfl(q, lo ? r : (r + 8), 32);

    float dmin[8];
    #pragma unroll
    for (int r = 0; r < 8; ++r) dmin[r] = 3.0e38f;

    const float* posb = pos + ((size_t)(b << 12)) * 3;

    for (int tt = 0; tt < 512; ++tt) {
        const int t  = tt * 16 + m;               // triangle index (N = m)
        const int i0 = faces[t * 3 + 0];
        const int i1 = faces[t * 3 + 1];
        const int i2 = faces[t * 3 + 2];
        const float* pa = posb + i0 * 3;
        const float* pb = posb + i1 * 3;
        const float* pq = posb + i2 * 3;
        const float ax = pa[0], ay = pa[1], az = pa[2];
        const float bx = pb[0], by = pb[1], bz = pb[2];
        const float cx = pq[0], cy = pq[1], cz = pq[2];

        const float abx = bx - ax, aby = by - ay, abz = bz - az;
        const float acx = cx - ax, acy = cy - ay, acz = cz - az;

        const float Aab = abx * ax + aby * ay + abz * az;
        const float Aac = acx * ax + acy * ay + acz * az;
        const float Bab = abx * bx + aby * by + abz * bz;
        const float Bac = acx * bx + acy * by + acz * bz;
        const float Cab = abx * cx + aby * cy + abz * cz;
        const float Cac = acx * cx + acy * cy + acz * cz;
        const float na2 = ax * ax + ay * ay + az * az;
        const float nb2 = bx * bx + by * by + bz * bz;
        const float nc2 = cx * cx + cy * cy + cz * cz;

        // B operands (4x16 K-major): VGPR0 = K0 / K2, VGPR1 = K1 / K3(=0)
        v2f Ba, Bb, Bc;
        Ba.x = lo ? ax : az;  Ba.y = lo ? ay : 0.0f;
        Bb.x = lo ? bx : bz;  Bb.y = lo ? by : 0.0f;
        Bc.x = lo ? cx : cz;  Bc.y = lo ? cy : 0.0f;

        v8f z = {};
        v8f Ga = __builtin_amdgcn_wmma_f32_16x16x4_f32(
            false, A, false, Ba, (short)0, z, false, false);
        v8f Gb = __builtin_amdgcn_wmma_f32_16x16x4_f32(
            false, A, false, Bb, (short)0, z, false, false);
        v8f Gc = __builtin_amdgcn_wmma_f32_16x16x4_f32(
            false, A, false, Bc, (short)0, z, false, false);

        #pragma unroll
        for (int r = 0; r < 8; ++r) {
            const float ga = Ga[r], gb = Gb[r], gc = Gc[r];
            const float gba = gb - ga;            // ab . p
            const float gca = gc - ga;            // ac . p
            const float d1 = gba - Aab;
            const float d2 = gca - Aac;
            const float d3 = gba - Bab;
            const float d4 = gca - Bac;
            const float d5 = gba - Cab;
            const float d6 = gca - Cac;
            const float qq  = pp[r];
            const float ap2 = qq - 2.0f * ga + na2;
            const float bp2 = qq - 2.0f * gb + nb2;
            const float cp2 = qq - 2.0f * gc + nc2;

            const float va = d3 * d6 - d5 * d4;
            const float vb = d5 * d2 - d1 * d6;
            const float vc = d1 * d4 - d3 * d2;
            const float denom = va + vb + vc;

            const float e1  = d1 - d3;            // |ab|^2
            const float e2  = d2 - d6;            // |ac|^2
            const float e12 = d1 - d5;            // ab.ac
            const float u    = d4 - d3;           // bc.bp
            const float dbc  = u + (d5 - d6);     // |bc|^2

            const float v  = sdiv_guard(vb, denom);
            const float w  = sdiv_guard(vc, denom);
            const float tab = sdiv_guard(d1, e1);
            const float tac = sdiv_guard(d2, e2);
            const float tbc = sdiv_guard(u, dbc);

            // interior
            float dist = ap2 - 2.0f * v * d1 - 2.0f * w * d2
                       + v * v * e1 + 2.0f * v * w * e12 + w * w * e2;
            // edge BC
            float dd = bp2 - 2.0f * tbc * u + tbc * tbc * dbc;
            dist = ((va <= 0.0f) && (u >= 0.0f) && ((d5 - d6) >= 0.0f)) ? dd : dist;
            // edge AC
            dd = ap2 - 2.0f * tac * d2 + tac * tac * e2;
            dist = ((vb <= 0.0f) && (d2 >= 0.0f) && (d6 <= 0.0f)) ? dd : dist;
            // edge AB
            dd = ap2 - 2.0f * tab * d1 + tab * tab * e1;
            dist = ((vc <= 0.0f) && (d1 >= 0.0f) && (d3 <= 0.0f)) ? dd : dist;
            // vertex C / B / A (highest priority last)
            dist = ((d6 >= 0.0f) && (d5 <= d6)) ? cp2 : dist;
            dist = ((d3 >= 0.0f) && (d4 <= d3)) ? bp2 : dist;
            dist = ((d1 <= 0.0f) && (d2 <= 0.0f)) ? ap2 : dist;

            dmin[r] = fminf(dmin[r], dist);
        }
    }

    // Min over N (16 triangles/lane-group); xor of bits 0..3 stays in-group.
    #pragma unroll
    for (int r = 0; r < 8; ++r) {
        #pragma unroll
        for (int off = 1; off < 16; off <<= 1)
            dmin[r] = fminf(dmin[r], __shfl_xor(dmin[r], off, 32));
    }

    // lane 0 owns points pbase+0..7, lane 16 owns pbase+8..15
    if (lane == 0 || lane == 16) {
        const int mb = pbase + ((lane == 0) ? 0 : 8);
        float s = 0.0f;
        #pragma unroll
        for (int r = 0; r < 8; ++r) {
            const float* q3 = pc + (((b << 12) + mb + r) * 3);
            const bool msk = (q3[0] != 0.0f) || (q3[1] != 0.0f) || (q3[2] != 0.0f);
            s += msk ? dmin[r] : 0.0f;
        }
        atomicAdd(&acc[b], s);
    }
}

// ---------------------------------------------------------------------------
// Kernel 3: masked counts per batch, mean over batches.
// ---------------------------------------------------------------------------
__global__ void mesh_finalize(const float* __restrict__ pc,
                              const float* __restrict__ acc,
                              float* __restrict__ out) {
    __shared__ float red[256];
    __shared__ float result;
    if (threadIdx.x == 0) result = 0.0f;
    __syncthreads();
    for (int b = 0; b < 4; ++b) {
        float partial = 0.0f;
        for (int i = threadIdx.x; i < 4096; i += 256) {
            const float* q3 = pc + ((b << 12) + i) * 3;
            partial += ((q3[0] != 0.0f) || (q3[1] != 0.0f) || (q3[2] != 0.0f))
                           ? 1.0f : 0.0f;
        }
        red[threadIdx.x] = partial;
        __syncthreads();
        for (int s = 128; s > 0; s >>= 1) {
            if (threadIdx.x < s) red[threadIdx.x] += red[threadIdx.x + s];
            __syncthreads();
        }
        if (threadIdx.x == 0) {
            const float cnt = fmaxf(red[0], 1.0f);
            result += acc[b] / cnt;
        }
        __syncthreads();
    }
    if (threadIdx.x == 0) out[0] = result * 0.25f;
}

// ---------------------------------------------------------------------------
extern "C" void kernel_launch(void* const* d_in, const int* in_sizes, int n_in,
                              void* d_out, int out_size, void* d_ws, size_t ws_size,
                              hipStream_t stream) {
    const float* verts = (const float*)d_in[0];   // (4,3,16,16,16) f32
    const float* pc    = (const float*)d_in[1];   // (4,4096,3)     f32
    const int*   faces = (const int*)d_in[2];     // (8192,3)       i32
    float*       out   = (float*)d_out;           // scalar f32

    float* acc = (float*)d_ws;                    // 4 floats
    float* pos = (float*)d_ws + 4;                // 4*4096*3 floats (~192KB)

    mesh_prep<<<64, 256, 0, stream>>>(verts, pos, acc);
    mesh_main<<<128, 256, 0, stream>>>(pos, pc, faces, acc);
    mesh_finalize<<<1, 256, 0, stream>>>(pc, acc, out);
}